// NodeNetwork_5403068859068
// MI455X (gfx1250) — compile-verified
//
#include <hip/hip_runtime.h>

typedef __attribute__((ext_vector_type(2))) float v2f;
typedef __attribute__((ext_vector_type(4))) float v4f;
typedef __attribute__((ext_vector_type(8))) float v8f;

#define NDEG   32
#define EMB    64
#define NMB    64
#define NFEAT  2
#define INDIM  130   // NMB + NFEAT + EMB
#define KPAD   132   // INDIM padded to multiple of 4
#define HID    128
#define TM     16    // nodes per block (WMMA M)

__global__ __launch_bounds__(256) void node_mlp_wmma(
    const float* __restrict__ mailbox,   // [N, 32, 64]
    const float* __restrict__ node_h,    // [N, 64]
    const float* __restrict__ node_f,    // [N, 2]
    const float* __restrict__ W1,        // [130, 128] row-major
    const float* __restrict__ b1,        // [128]
    const float* __restrict__ W2,        // [128, 64] row-major
    const float* __restrict__ b2,        // [64]
    float* __restrict__ out,             // [N, 64]
    int nNodes)
{
    __shared__ float Xs[TM * KPAD];      // concat input tile [16][132]
    __shared__ float Hs[TM * KPAD];      // hidden tile       [16][132] (cols 0..127 used)

    const int t    = threadIdx.x;
    const int base = blockIdx.x * TM;

    // ---------------- Stage A: mailbox aggregation + concat into LDS -------------
    {
        const int m   = t >> 4;          // node in tile 0..15
        const int f4  = (t & 15) * 4;    // feature quad 0..60
        const int node = base + m;
        v4f acc = {0.f, 0.f, 0.f, 0.f};
        v4f nh  = {0.f, 0.f, 0.f, 0.f};
        if (node < nNodes) {
            const v4f* mb = (const v4f*)(mailbox + (size_t)node * (NDEG * EMB) + f4);
            #pragma unroll 8
            for (int msg = 0; msg < NDEG; ++msg) {
                // streaming, read-once: non-temporal so W1/W2 stay hot in L2
                v4f v = __builtin_nontemporal_load(&mb[msg * (EMB / 4)]);
                acc += v;
            }
            nh = *(const v4f*)(node_h + (size_t)node * NMB + f4);
        }
        // cols [0,64): node hidden state
        Xs[m * KPAD + f4 + 0] = nh.x;
        Xs[m * KPAD + f4 + 1] = nh.y;
        Xs[m * KPAD + f4 + 2] = nh.z;
        Xs[m * KPAD + f4 + 3] = nh.w;
        // cols [66,130): aggregated edge messages
        Xs[m * KPAD + 66 + f4 + 0] = acc.x;
        Xs[m * KPAD + 66 + f4 + 1] = acc.y;
        Xs[m * KPAD + 66 + f4 + 2] = acc.z;
        Xs[m * KPAD + 66 + f4 + 3] = acc.w;
    }
    if (t < 32) {                        // cols [64,66): raw node features
        const int m = t >> 1, f = t & 1;
        const int node = base + m;
        Xs[m * KPAD + NMB + f] = (node < nNodes) ? node_f[(size_t)node * NFEAT + f] : 0.f;
    } else if (t < 64) {                 // cols [130,132): zero pad
        const int m = (t - 32) >> 1, f = (t - 32) & 1;
        Xs[m * KPAD + INDIM + f] = 0.f;
    }
    __syncthreads();

    const int lane = t & 31;
    const int wv   = t >> 5;             // wave 0..7
    const int mrow = lane & 15;          // A-matrix M row for this lane
    const int koff = (lane >> 4) << 1;   // lanes 0-15 -> K{0,1}, lanes 16-31 -> K{2,3}
    const int rofs = (lane >> 4) << 3;   // C/D: lanes 16-31 hold rows M+8

    // ---------------- Stage B: h = relu(X @ W1 + b1), 8 waves x 16 cols ----------
    {
        const int nn = wv * 16 + (lane & 15);    // output column in [0,128)
        v8f acc = {};
        #pragma unroll 4
        for (int k0 = 0; k0 < KPAD; k0 += 4) {
            const int kk = k0 + koff;
            v2f a, b;
            a.x = Xs[mrow * KPAD + kk];
            a.y = Xs[mrow * KPAD + kk + 1];
            b.x = (kk     < INDIM) ? W1[(size_t)kk * HID + nn]       : 0.f;
            b.y = (kk + 1 < INDIM) ? W1[(size_t)(kk + 1) * HID + nn] : 0.f;
            acc = __builtin_amdgcn_wmma_f32_16x16x4_f32(
                      false, a, false, b, (short)0, acc, false, false);
        }
        const float bias = b1[nn];
        #pragma unroll
        for (int r = 0; r < 8; ++r) {
            const float v = acc[r] + bias;
            Hs[(r + rofs) * KPAD + nn] = v > 0.f ? v : 0.f;
        }
    }
    __syncthreads();

    // ---------------- Stage C: out = relu(H @ W2 + b2), 4 waves x 16 cols --------
    if (wv < 4) {
        const int nn = wv * 16 + (lane & 15);    // output column in [0,64)
        v8f acc = {};
        #pragma unroll 4
        for (int k0 = 0; k0 < HID; k0 += 4) {
            const int kk = k0 + koff;
            v2f a, b;
            a.x = Hs[mrow * KPAD + kk];
            a.y = Hs[mrow * KPAD + kk + 1];
            b.x = W2[(size_t)kk * NMB + nn];
            b.y = W2[(size_t)(kk + 1) * NMB + nn];
            acc = __builtin_amdgcn_wmma_f32_16x16x4_f32(
                      false, a, false, b, (short)0, acc, false, false);
        }
        const float bias = b2[nn];
        #pragma unroll
        for (int r = 0; r < 8; ++r) {
            const int node = base + r + rofs;
            if (node < nNodes) {
                const float v = acc[r] + bias;
                out[(size_t)node * NMB + nn] = v > 0.f ? v : 0.f;
            }
        }
    }
}

extern "C" void kernel_launch(void* const* d_in, const int* in_sizes, int n_in,
                              void* d_out, int out_size, void* d_ws, size_t ws_size,
                              hipStream_t stream) {
    const float* mailbox = (const float*)d_in[0];
    const float* node_h  = (const float*)d_in[1];
    const float* node_f  = (const float*)d_in[2];
    const float* W1      = (const float*)d_in[3];
    const float* b1      = (const float*)d_in[4];
    const float* W2      = (const float*)d_in[5];
    const float* b2      = (const float*)d_in[6];
    float* out = (float*)d_out;

    const int nNodes = in_sizes[2] / NFEAT;      // node_features is [N,2]
    const int blocks = (nNodes + TM - 1) / TM;
    node_mlp_wmma<<<blocks, 256, 0, stream>>>(mailbox, node_h, node_f,
                                              W1, b1, W2, b2, out, nNodes);
}